// MultiHeadGraphAttention_30331059044543
// MI455X (gfx1250) — compile-verified
//
#include <hip/hip_runtime.h>

// ---------------------------------------------------------------------------
// MultiHeadGraphAttention on MI455X (gfx1250, wave32)
//   N=50000 nodes, E=600000 edges, D=128, H=8, DH=16, SCALE=4
// GEMMs via V_WMMA_F32_16X16X4_F32 (fp32 in/out, memory-bound -> keep fp32).
// Edge phase is L2-resident (Q+K+V+accum = 102 MB < 192 MB L2).
// ---------------------------------------------------------------------------

constexpr int   kN        = 50000;
constexpr int   kE        = 600000;
constexpr int   kD        = 128;
constexpr int   kH        = 8;
constexpr float kInvScale = 0.25f;   // 1/sqrt(DH) = 1/4

typedef float v2f __attribute__((ext_vector_type(2)));
typedef float v8f __attribute__((ext_vector_type(8)));

// Monotone float <-> uint mapping so atomicMax(uint) == float max.
__device__ __forceinline__ unsigned f2ord(float f) {
  unsigned u = __float_as_uint(f);
  return u ^ ((u >> 31) ? 0xFFFFFFFFu : 0x80000000u);
}
__device__ __forceinline__ float ord2f(unsigned u) {
  unsigned b = (u & 0x80000000u) ? (u ^ 0x80000000u) : ~u;
  return __uint_as_float(b);
}

// ---------------------------------------------------------------------------
// GEMM: O[16*gridDim.x x 128] = A[... x 128] @ W[128 x 128] + bias
// 256 threads = 8 waves; wave w computes one 16x16 tile (col tile w) with a
// chain of 32 v_wmma_f32_16x16x4_f32.
// A frag (16x4 f32): lane L -> m = L&15, ks = 2*(L>>4); holds K=ks,ks+1.
// B frag (4x16 f32): lane L -> n = L&15, same K split.
// C/D (16x16 f32):   VGPR r -> M = r + 8*(L>>4), N = L&15.
// ---------------------------------------------------------------------------
__global__ void node_gemm_16x16x4(const float* __restrict__ A,
                                  const float* __restrict__ W,
                                  const float* __restrict__ bias,
                                  float* __restrict__ O) {
  const int lane = threadIdx.x & 31;
  const int wave = threadIdx.x >> 5;           // 0..7 : 16-col tile
  const int m    = lane & 15;
  const int ksub = (lane >> 4) << 1;           // 0 or 2
  const int row  = blockIdx.x * 16 + m;
  const int col  = wave * 16 + m;              // n index (lane&15)

  const float* __restrict__ arow = A + (size_t)row * kD;

  v8f c = {0.f, 0.f, 0.f, 0.f, 0.f, 0.f, 0.f, 0.f};
#pragma unroll
  for (int k0 = 0; k0 < kD; k0 += 4) {
    v2f a, b;
    a.x = arow[k0 + ksub];
    a.y = arow[k0 + ksub + 1];
    b.x = W[(size_t)(k0 + ksub) * kD + col];
    b.y = W[(size_t)(k0 + ksub + 1) * kD + col];
    c = __builtin_amdgcn_wmma_f32_16x16x4_f32(false, a, false, b, (short)0, c,
                                              false, false);
  }

  const float bv    = bias[col];
  const int   mbase = blockIdx.x * 16 + ((lane >> 4) << 3);
#pragma unroll
  for (int r = 0; r < 8; ++r) {
    O[(size_t)(mbase + r) * kD + col] = c[r] + bv;
  }
}

// ---------------------------------------------------------------------------
// Pass 1: scores[e][h] = <Q[row[e],h,:], K[col[e],h,:]>/4 + edge_attr[e];
// per-edge max over 8 heads via lane shuffles, then ordered-int atomicMax
// into node_max[row]. Thread t = e*8 + h  => 8 lanes of one edge are
// contiguous, so the 512B Q/K row loads coalesce.
// ---------------------------------------------------------------------------
__global__ void edge_scores_kernel(const float* __restrict__ Q,
                                   const float* __restrict__ K,
                                   const int* __restrict__ row,
                                   const int* __restrict__ col,
                                   const float* __restrict__ eattr,
                                   float* __restrict__ scores,
                                   unsigned* __restrict__ node_max) {
  const int t = blockIdx.x * blockDim.x + threadIdx.x;
  if (t >= kE * kH) return;
  const int e = t >> 3, h = t & 7;
  const int r = row[e], cj = col[e];

  const float4* q = (const float4*)(Q + (size_t)r * kD + h * 16);
  const float4* k = (const float4*)(K + (size_t)cj * kD + h * 16);
  float s = 0.f;
#pragma unroll
  for (int i = 0; i < 4; ++i) {
    float4 qv = q[i], kv = k[i];
    s += qv.x * kv.x + qv.y * kv.y + qv.z * kv.z + qv.w * kv.w;
  }
  s = s * kInvScale + eattr[e];
  scores[t] = s;

  // max over the 8 heads of this edge (lanes t..t|7 within the wave32)
  float mx = s;
  mx = fmaxf(mx, __shfl_xor(mx, 4, 32));
  mx = fmaxf(mx, __shfl_xor(mx, 2, 32));
  mx = fmaxf(mx, __shfl_xor(mx, 1, 32));
  if (h == 0) atomicMax(node_max + r, f2ord(mx));
}

// Pass 2: s_exp = exp(score - node_max[row]); segment-sum per (row, head).
__global__ void edge_exp_kernel(float* __restrict__ scores,
                                const unsigned* __restrict__ node_max,
                                const int* __restrict__ row,
                                float* __restrict__ s_sum) {
  const int t = blockIdx.x * blockDim.x + threadIdx.x;
  if (t >= kE * kH) return;
  const int e = t >> 3, h = t & 7;
  const int r = row[e];
  const float sx = __expf(scores[t] - ord2f(node_max[r]));
  scores[t] = sx;
  atomicAdd(s_sum + (size_t)r * kH + h, sx);
}

// Pass 3: accum[row, h, :] += s_exp * V[col, h, :]  (unnormalized; the
// per-(row,head) 1/(sum+eps) factor is applied afterwards, exactly).
__global__ void edge_scatter_kernel(const float* __restrict__ scores,
                                    const float* __restrict__ V,
                                    const int* __restrict__ row,
                                    const int* __restrict__ col,
                                    float* __restrict__ accum) {
  const int t = blockIdx.x * blockDim.x + threadIdx.x;
  if (t >= kE * kH) return;
  const int e = t >> 3, h = t & 7;
  const int r = row[e], cj = col[e];
  const float w = scores[t];
  const float4* v = (const float4*)(V + (size_t)cj * kD + h * 16);
  float* a = accum + (size_t)r * kD + h * 16;
#pragma unroll
  for (int i = 0; i < 4; ++i) {
    float4 vv = v[i];
    atomicAdd(a + 4 * i + 0, w * vv.x);
    atomicAdd(a + 4 * i + 1, w * vv.y);
    atomicAdd(a + 4 * i + 2, w * vv.z);
    atomicAdd(a + 4 * i + 3, w * vv.w);
  }
}

// accum[m][k] *= 1/(s_sum[m][k/16] + 1e-8)
__global__ void normalize_kernel(float* __restrict__ accum,
                                 const float* __restrict__ s_sum) {
  const int t = blockIdx.x * blockDim.x + threadIdx.x;
  if (t >= kN * kD) return;
  const int m = t >> 7;
  const int h = (t >> 4) & 7;
  accum[t] *= 1.0f / (s_sum[(size_t)m * kH + h] + 1e-8f);
}

__global__ void init_stats_kernel(float* __restrict__ s_sum,
                                  unsigned* __restrict__ node_max) {
  const int t = blockIdx.x * blockDim.x + threadIdx.x;
  if (t < kN * kH) s_sum[t] = 0.f;
  if (t < kN) node_max[t] = 0x007FFFFFu;   // f2ord(-inf)
}

__global__ void zero_kernel(float* __restrict__ p, int n) {
  const int t = blockIdx.x * blockDim.x + threadIdx.x;
  if (t < n) p[t] = 0.f;
}

// ---------------------------------------------------------------------------
extern "C" void kernel_launch(void* const* d_in, const int* in_sizes, int n_in,
                              void* d_out, int out_size, void* d_ws,
                              size_t ws_size, hipStream_t stream) {
  (void)in_sizes; (void)n_in; (void)out_size; (void)ws_size;

  const float* x  = (const float*)d_in[0];
  const int*   ei = (const int*)d_in[1];     // [2, E] int
  const float* ea = (const float*)d_in[2];
  const float* Wq = (const float*)d_in[3];
  const float* bq = (const float*)d_in[4];
  const float* Wk = (const float*)d_in[5];
  const float* bk = (const float*)d_in[6];
  const float* Wv = (const float*)d_in[7];
  const float* bv = (const float*)d_in[8];
  const float* Wo = (const float*)d_in[9];
  const float* bo = (const float*)d_in[10];
  const int* row = ei;
  const int* col = ei + kE;
  float* out = (float*)d_out;

  // Workspace carve-up (256B aligned). accum reuses the Q buffer after the
  // score pass (Q dead after edge_scores).
  char*  ws  = (char*)d_ws;
  size_t off = 0;
  auto carve = [&](size_t bytes) {
    char* p = ws + off;
    off += (bytes + 255) & ~size_t(255);
    return p;
  };
  float*    Qb       = (float*)carve((size_t)kN * kD * 4);
  float*    Kb       = (float*)carve((size_t)kN * kD * 4);
  float*    Vb       = (float*)carve((size_t)kN * kD * 4);
  float*    scores   = (float*)carve((size_t)kE * kH * 4);
  float*    s_sum    = (float*)carve((size_t)kN * kH * 4);
  unsigned* node_max = (unsigned*)carve((size_t)kN * 4);
  float*    accum    = Qb;

  const int rowTiles = kN / 16;   // 3125, exact
  const int eh       = kE * kH;   // 4.8M

  init_stats_kernel<<<(kN * kH + 255) / 256, 256, 0, stream>>>(s_sum, node_max);

  node_gemm_16x16x4<<<rowTiles, 256, 0, stream>>>(x, Wq, bq, Qb);
  node_gemm_16x16x4<<<rowTiles, 256, 0, stream>>>(x, Wk, bk, Kb);
  node_gemm_16x16x4<<<rowTiles, 256, 0, stream>>>(x, Wv, bv, Vb);

  edge_scores_kernel<<<(eh + 255) / 256, 256, 0, stream>>>(
      Qb, Kb, row, col, ea, scores, node_max);

  zero_kernel<<<(kN * kD + 255) / 256, 256, 0, stream>>>(accum, kN * kD);

  edge_exp_kernel<<<(eh + 255) / 256, 256, 0, stream>>>(scores, node_max, row,
                                                        s_sum);
  edge_scatter_kernel<<<(eh + 255) / 256, 256, 0, stream>>>(scores, Vb, row,
                                                            col, accum);
  normalize_kernel<<<(kN * kD + 255) / 256, 256, 0, stream>>>(accum, s_sum);

  node_gemm_16x16x4<<<rowTiles, 256, 0, stream>>>(accum, Wo, bo, out);
}